// NSELoss_7301444403966
// MI455X (gfx1250) — compile-verified
//
#include <hip/hip_runtime.h>

// Problem constants (match reference setup_inputs)
#define T_DIM   2000
#define G_DIM   8192
#define NCHUNK  10
#define TCHUNK  (T_DIM / NCHUNK)   // 200 timesteps per chunk
#define BLK     256
#define GBLK    (G_DIM / BLK)      // 32 gage blocks

// ---------------------------------------------------------------------------
// CDNA5 async global->LDS path (guarded; falls back to plain copy)
// ---------------------------------------------------------------------------
#if defined(__HIP_DEVICE_COMPILE__) && __has_builtin(__builtin_amdgcn_global_load_async_to_lds_b128)
#define USE_ASYNC 1
#else
#define USE_ASYNC 0
#endif

// Match the builtin's parameter types exactly (from hipcc diagnostic):
//   param0: __attribute__((vector_size(16))) int  AS(1)*   (global source)
//   param1: presumed same vector type in AS(3)             (LDS destination)
typedef int v4i_gcc __attribute__((vector_size(16)));
typedef __attribute__((address_space(1))) v4i_gcc* g_v4i_p;
typedef __attribute__((address_space(3))) v4i_gcc* l_v4i_p;

__device__ __forceinline__ void ld16_to_lds(const float4* gp, float4* lp) {
#if USE_ASYNC
    // 16 bytes/lane, DMA into LDS, tracked by ASYNCcnt
    __builtin_amdgcn_global_load_async_to_lds_b128(
        (g_v4i_p)(float4*)gp,   // drop const, reinterpret, addrspacecast -> AS(1)
        (l_v4i_p)lp,            // generic -> AS(3)
        /*offset=*/0, /*cpol=*/0);
#else
    *lp = *gp;
#endif
}

template <int N>
__device__ __forceinline__ void wait_async_le() {
#if USE_ASYNC
#if __has_builtin(__builtin_amdgcn_s_wait_asynccnt)
    __builtin_amdgcn_s_wait_asynccnt(N);
#else
    asm volatile("s_wait_asynccnt %0" ::"i"(N) : "memory");
#endif
#endif
}

// ---------------------------------------------------------------------------
// Stage 1: masked one-pass partial stats per (gage, k) over a T-chunk.
// Per (chunk, gage): 4 x float4 {cnt, sum_t, sum_t2, ssres} written to ws.
// ---------------------------------------------------------------------------
__global__ __launch_bounds__(BLK) void nse_stage1(const float4* __restrict__ pred,
                                                  const float4* __restrict__ targ,
                                                  float4* __restrict__ part) {
    __shared__ float4 sP[2][BLK];
    __shared__ float4 sT[2][BLK];

    const int tid = threadIdx.x;
    const int g   = blockIdx.x * BLK + tid;   // gage index
    const int t0  = blockIdx.y * TCHUNK;      // chunk start

    float cnt[4] = {0.f, 0.f, 0.f, 0.f};
    float s1[4]  = {0.f, 0.f, 0.f, 0.f};
    float s2[4]  = {0.f, 0.f, 0.f, 0.f};
    float sr[4]  = {0.f, 0.f, 0.f, 0.f};

    auto acc1 = [&](float tv, float pv, int k) {
        const bool  m = !__builtin_isnan(tv);
        const float t = m ? tv : 0.f;
        const float p = m ? pv : 0.f;
        cnt[k] += m ? 1.f : 0.f;
        s1[k]  += t;
        s2[k]   = fmaf(t, t, s2[k]);
        const float d = t - p;
        sr[k]   = fmaf(d, d, sr[k]);
    };

    auto issue = [&](int i, int b) {
        const size_t off = (size_t)(t0 + i) * G_DIM + g;
        ld16_to_lds(pred + off, &sP[b][tid]);
        ld16_to_lds(targ + off, &sT[b][tid]);
    };

    auto consume = [&](int b) {
        const float4 tv = sT[b][tid];
        const float4 pv = sP[b][tid];
        acc1(tv.x, pv.x, 0);
        acc1(tv.y, pv.y, 1);
        acc1(tv.z, pv.z, 2);
        acc1(tv.w, pv.w, 3);
    };

    // Double-buffered pipeline: 2 async loads in flight per stage.
    issue(0, 0);
    for (int i = 0; i < TCHUNK - 1; ++i) {
        issue(i + 1, (i + 1) & 1);
        wait_async_le<2>();     // oldest pair (buffer i&1) has landed in LDS
        consume(i & 1);
    }
    wait_async_le<0>();
    consume((TCHUNK - 1) & 1);

    const size_t base = ((size_t)blockIdx.y * G_DIM + g) * 4;
#pragma unroll
    for (int k = 0; k < 4; ++k)
        part[base + k] = make_float4(cnt[k], s1[k], s2[k], sr[k]);
}

// ---------------------------------------------------------------------------
// Stage 2: combine chunks, per-gage NSE + validity, block-reduce per k.
// blockout[b*8 + k]   = sum of NSE over this block's gages (k = 0..3)
// blockout[b*8 + 4+k] = count of valid gages
// ---------------------------------------------------------------------------
__global__ __launch_bounds__(BLK) void nse_stage2(const float4* __restrict__ part,
                                                  float* __restrict__ blockout) {
    const int tid = threadIdx.x;
    const int g   = blockIdx.x * BLK + tid;

    float a[4][4];
#pragma unroll
    for (int k = 0; k < 4; ++k)
#pragma unroll
        for (int j = 0; j < 4; ++j) a[k][j] = 0.f;

    for (int c = 0; c < NCHUNK; ++c) {
        const size_t base = ((size_t)c * G_DIM + g) * 4;
#pragma unroll
        for (int k = 0; k < 4; ++k) {
            const float4 v = part[base + k];
            a[k][0] += v.x; a[k][1] += v.y; a[k][2] += v.z; a[k][3] += v.w;
        }
    }

    __shared__ float red[8][BLK];
#pragma unroll
    for (int k = 0; k < 4; ++k) {
        const float cnt = a[k][0], s1 = a[k][1], s2 = a[k][2], sr = a[k][3];
        const float sc   = fmaxf(cnt, 1.f);
        const float mean = s1 / sc;
        const float sst  = s2 - s1 * mean;          // == sum (t-mean)^2 on masked set
        const bool  valid = (cnt > 0.f) && (sst != 0.f);
        const float nse   = valid ? (1.f - sr / sst) : 0.f;
        red[k][tid]     = nse;
        red[4 + k][tid] = valid ? 1.f : 0.f;
    }
    __syncthreads();

    for (int s = BLK / 2; s > 0; s >>= 1) {
        if (tid < s) {
#pragma unroll
            for (int q = 0; q < 8; ++q) red[q][tid] += red[q][tid + s];
        }
        __syncthreads();
    }
    if (tid < 8) blockout[blockIdx.x * 8 + tid] = red[tid][0];
}

// ---------------------------------------------------------------------------
// Stage 3: final scalar = sum_k -( losssum_k / nsample_k )
// ---------------------------------------------------------------------------
__global__ void nse_stage3(const float* __restrict__ blockout, float* __restrict__ out) {
    if (blockIdx.x == 0 && threadIdx.x == 0) {
        float loss[4] = {0.f, 0.f, 0.f, 0.f};
        float n[4]    = {0.f, 0.f, 0.f, 0.f};
        for (int b = 0; b < GBLK; ++b) {
#pragma unroll
            for (int k = 0; k < 4; ++k) {
                loss[k] += blockout[b * 8 + k];
                n[k]    += blockout[b * 8 + 4 + k];
            }
        }
        float r = 0.f;
#pragma unroll
        for (int k = 0; k < 4; ++k) r -= loss[k] / n[k];
        out[0] = r;
    }
}

// ---------------------------------------------------------------------------
extern "C" void kernel_launch(void* const* d_in, const int* in_sizes, int n_in,
                              void* d_out, int out_size, void* d_ws, size_t ws_size,
                              hipStream_t stream) {
    (void)in_sizes; (void)n_in; (void)out_size; (void)ws_size;

    const float4* pred = (const float4*)d_in[0];   // "output"
    const float4* targ = (const float4*)d_in[1];   // "target" (NaN = missing)

    float4* part     = (float4*)d_ws;              // NCHUNK*G_DIM*4 float4 = 5.25 MB
    float*  blockout = (float*)((char*)d_ws + (size_t)NCHUNK * G_DIM * 4 * sizeof(float4));
    float*  out      = (float*)d_out;

    dim3 g1(GBLK, NCHUNK);
    nse_stage1<<<g1, BLK, 0, stream>>>(pred, targ, part);
    nse_stage2<<<GBLK, BLK, 0, stream>>>(part, blockout);
    nse_stage3<<<1, 64, 0, stream>>>(blockout, out);
}